// EncoderDecoder_61220463837553
// MI455X (gfx1250) — compile-verified
//
#include <hip/hip_runtime.h>
#include <math.h>

// ---------------------------------------------------------------------------
// GRU encoder-decoder, H=1024, N=16384, fp32 end-to-end.
// Heavy op per step: gh = h @ Whh.T  (16384x1024 @ 1024x3072) fused with the
// elementwise GRU update. Uses native CDNA5 fp32 WMMA (V_WMMA_F32_16X16X4_F32).
// v2: swizzled LDS layout -> pure ds_load_b64 fragment feeds (no cndmask),
//     software-pipelined global->LDS staging (register double buffer).
// ---------------------------------------------------------------------------

#define GRU_N 16384
#define GRU_H 1024

typedef float v8f __attribute__((ext_vector_type(8)));
typedef float v2f __attribute__((ext_vector_type(2)));

__device__ __forceinline__ float sigmoid_f(float x) {
    return 1.0f / (1.0f + __expf(-x));
}

// ---------------------------------------------------------------------------
// Step 0 (encoder, h == 0): gh collapses to bhh, pure elementwise.
// ---------------------------------------------------------------------------
__global__ void gru_step0_kernel(float* __restrict__ h_out,
                                 const float* __restrict__ X,   // stride 8
                                 const float* __restrict__ Wih, // (3H, 2)
                                 const float* __restrict__ bih,
                                 const float* __restrict__ bhh)
{
    int idx = blockIdx.x * 256 + threadIdx.x;          // 0 .. N*H-1
    int n = idx >> 10;
    int j = idx & (GRU_H - 1);
    float x0 = X[n * 8 + 0];
    float x1 = X[n * 8 + 1];
    int jr = j, jz = j + GRU_H, jn = j + 2 * GRU_H;
    float ir  = fmaf(x0, Wih[2 * jr], fmaf(x1, Wih[2 * jr + 1], bih[jr]));
    float iz  = fmaf(x0, Wih[2 * jz], fmaf(x1, Wih[2 * jz + 1], bih[jz]));
    float in_ = fmaf(x0, Wih[2 * jn], fmaf(x1, Wih[2 * jn + 1], bih[jn]));
    float r  = sigmoid_f(ir + bhh[jr]);
    float z  = sigmoid_f(iz + bhh[jz]);
    float nn = tanhf(fmaf(r, bhh[jn], in_));
    h_out[idx] = (1.0f - z) * nn;                      // h_prev == 0
}

// ---------------------------------------------------------------------------
// Fused GRU step. Block: 256 threads (8 wave32). BM=64 rows x BN=64 h-cols
// -> 64x192 gate tile, all three gates per wave so the update is in-register.
//
// LDS fragment layout (f32 WMMA 16x16x4, ISA 7.12.2):
//   A (16x4):  lane<16 needs (k,k+1), lane>=16 needs (k+2,k+3)  -> natural
//              order, ds_load_b64 at k4 + 2*hi.
//   B (4x16):  lane<16 needs (k,k+2), lane>=16 needs (k+1,k+3)  -> store the
//              W tile permuted per k-group of 4 as (k, k+2, k+1, k+3), then
//              ds_load_b64 at k4 + 2*hi.
// ---------------------------------------------------------------------------
#define BM 64
#define BN 64
#define KC 32
#define LDP 36   // padded LDS row stride in floats (144 B = 9*16)

__launch_bounds__(256)
__global__ void gru_step_kernel(const float* __restrict__ h_in,
                                float* __restrict__ h_out,
                                const float* __restrict__ x, int x_stride,
                                const float* __restrict__ Whh,  // (3H, H)
                                const float* __restrict__ Wih,  // (3H, 2)
                                const float* __restrict__ bih,
                                const float* __restrict__ bhh)
{
    __shared__ __align__(16) float lds_h[BM * LDP];        //  9.2 KB
    __shared__ __align__(16) float lds_w[192 * LDP];       // 27.6 KB
    __shared__ float x_s[BM * 2];
    __shared__ float bih_s[192], bhh_s[192], wih_s[192 * 2];

    const int tid  = threadIdx.x;
    const int lane = tid & 31;
    const int wv   = tid >> 5;
    const int row0 = blockIdx.x * BM;
    const int c0   = blockIdx.y * BN;

    // ---- stage per-block x, biases, Wih columns ---------------------------
    if (tid < 128) {
        int r = tid >> 1, o = tid & 1;
        x_s[tid] = x[(row0 + r) * x_stride + o];
    }
    for (int g = tid; g < 192; g += 256) {
        int gcol = (g >> 6) * GRU_H + c0 + (g & 63);       // global gate column
        bih_s[g] = bih[gcol];
        bhh_s[g] = bhh[gcol];
        wih_s[2 * g]     = Wih[2 * gcol];
        wih_s[2 * g + 1] = Wih[2 * gcol + 1];
    }

    // kt-invariant loader coordinates
    const int hA_r = (tid * 4) >> 5;           // h-tile pass 0 row
    const int hA_c = (tid * 4) & 31;
    const int hB_r = (tid * 4 + 1024) >> 5;    // h-tile pass 1 row
    const int hB_c = (tid * 4 + 1024) & 31;
    int  w_g[6], w_c[6], w_row[6];
    #pragma unroll
    for (int i = 0; i < 6; ++i) {
        int lin = tid * 4 + i * 1024;
        w_g[i] = lin >> 5;
        w_c[i] = lin & 31;
        w_row[i] = (w_g[i] >> 6) * GRU_H + c0 + (w_g[i] & 63);
    }

    const int  m0 = (wv & 3) * 16;       // wave row-tile base (local)
    const int  n0 = (wv >> 2) * 32;      // wave h-col base (local)
    const int  ln = lane & 15;
    const bool hi = lane >= 16;
    const int  fsel = hi ? 2 : 0;        // fragment half offset (floats)

    v8f acc[3][2];
    #pragma unroll
    for (int g = 0; g < 3; ++g)
        #pragma unroll
        for (int cc = 0; cc < 2; ++cc)
            acc[g][cc] = (v8f){0.f, 0.f, 0.f, 0.f, 0.f, 0.f, 0.f, 0.f};

    // ---- prologue: prefetch tile kt=0 into registers ----------------------
    float4 ph0, ph1, pw[6];
    ph0 = *(const float4*)&h_in[(size_t)(row0 + hA_r) * GRU_H + hA_c];
    ph1 = *(const float4*)&h_in[(size_t)(row0 + hB_r) * GRU_H + hB_c];
    #pragma unroll
    for (int i = 0; i < 6; ++i)
        pw[i] = *(const float4*)&Whh[(size_t)w_row[i] * GRU_H + w_c[i]];

    // ---- K loop (software pipelined) --------------------------------------
    for (int kt = 0; kt < GRU_H; kt += KC) {
        __syncthreads();   // previous compute done before LDS overwrite
        *(float4*)&lds_h[hA_r * LDP + hA_c] = ph0;
        *(float4*)&lds_h[hB_r * LDP + hB_c] = ph1;
        #pragma unroll
        for (int i = 0; i < 6; ++i) {
            // permute k-group (k,k+1,k+2,k+3) -> (k,k+2,k+1,k+3) for B frags
            float4 v = pw[i];
            *(float4*)&lds_w[w_g[i] * LDP + w_c[i]] = float4{v.x, v.z, v.y, v.w};
        }
        __syncthreads();

        if (kt + KC < GRU_H) {     // issue next tile's global loads now;
            int kn = kt + KC;      // latency hidden under the WMMA phase
            ph0 = *(const float4*)&h_in[(size_t)(row0 + hA_r) * GRU_H + kn + hA_c];
            ph1 = *(const float4*)&h_in[(size_t)(row0 + hB_r) * GRU_H + kn + hB_c];
            #pragma unroll
            for (int i = 0; i < 6; ++i)
                pw[i] = *(const float4*)&Whh[(size_t)w_row[i] * GRU_H + kn + w_c[i]];
        }

        #pragma unroll
        for (int k4 = 0; k4 < KC; k4 += 4) {
            v2f a = *(const v2f*)&lds_h[(m0 + ln) * LDP + k4 + fsel];
            #pragma unroll
            for (int g = 0; g < 3; ++g) {
                #pragma unroll
                for (int cc = 0; cc < 2; ++cc) {
                    v2f b = *(const v2f*)
                        &lds_w[(g * 64 + n0 + cc * 16 + ln) * LDP + k4 + fsel];
                    acc[g][cc] = __builtin_amdgcn_wmma_f32_16x16x4_f32(
                        false, a, false, b, (short)0, acc[g][cc], false, false);
                }
            }
        }
    }

    // ---- fused GRU epilogue (all three gates in-register) -----------------
    #pragma unroll
    for (int cc = 0; cc < 2; ++cc) {
        #pragma unroll
        for (int v = 0; v < 8; ++v) {
            int Rl = m0 + v + (hi ? 8 : 0);        // local row 0..63
            int Cl = n0 + cc * 16 + ln;            // local h-col 0..63
            int gr = Cl, gz = 64 + Cl, gn = 128 + Cl;
            float x0 = x_s[2 * Rl], x1 = x_s[2 * Rl + 1];
            float ir  = fmaf(x0, wih_s[2 * gr], fmaf(x1, wih_s[2 * gr + 1], bih_s[gr]));
            float iz  = fmaf(x0, wih_s[2 * gz], fmaf(x1, wih_s[2 * gz + 1], bih_s[gz]));
            float in_ = fmaf(x0, wih_s[2 * gn], fmaf(x1, wih_s[2 * gn + 1], bih_s[gn]));
            float hr = acc[0][cc][v] + bhh_s[gr];
            float hz = acc[1][cc][v] + bhh_s[gz];
            float hn = acc[2][cc][v] + bhh_s[gn];
            float r  = sigmoid_f(ir + hr);
            float z  = sigmoid_f(iz + hz);
            float nn = tanhf(fmaf(r, hn, in_));
            size_t off = (size_t)(row0 + Rl) * GRU_H + c0 + Cl;
            float hp = h_in[off];
            h_out[off] = fmaf(z, hp - nn, nn);      // (1-z)*nn + z*hp
        }
    }
}

// ---------------------------------------------------------------------------
// Output projection: out[n, 0..1] = h[n,:] @ out_W.T + out_b. One wave / row,
// b128 per-lane loads (128 floats per wave per pass).
// ---------------------------------------------------------------------------
__global__ void proj_kernel(const float* __restrict__ h,
                            const float* __restrict__ out_W, // (2, H)
                            const float* __restrict__ out_b,
                            float* __restrict__ out)         // stride 4, 2 cols
{
    int lane = threadIdx.x & 31;
    int wv   = threadIdx.x >> 5;
    int row  = blockIdx.x * 8 + wv;
    const float* hr = h + (size_t)row * GRU_H;
    float a0 = 0.f, a1 = 0.f;
    #pragma unroll
    for (int i = 0; i < GRU_H; i += 128) {
        float4 hv = *(const float4*)&hr[i + lane * 4];
        float4 w0 = *(const float4*)&out_W[i + lane * 4];
        float4 w1 = *(const float4*)&out_W[GRU_H + i + lane * 4];
        a0 = fmaf(hv.x, w0.x, fmaf(hv.y, w0.y, fmaf(hv.z, w0.z, fmaf(hv.w, w0.w, a0))));
        a1 = fmaf(hv.x, w1.x, fmaf(hv.y, w1.y, fmaf(hv.z, w1.z, fmaf(hv.w, w1.w, a1))));
    }
    #pragma unroll
    for (int off = 16; off > 0; off >>= 1) {
        a0 += __shfl_down(a0, off, 32);
        a1 += __shfl_down(a1, off, 32);
    }
    if (lane == 0) {
        out[(size_t)row * 4 + 0] = a0 + out_b[0];
        out[(size_t)row * 4 + 1] = a1 + out_b[1];
    }
}

// ---------------------------------------------------------------------------
extern "C" void kernel_launch(void* const* d_in, const int* in_sizes, int n_in,
                              void* d_out, int out_size, void* d_ws, size_t ws_size,
                              hipStream_t stream)
{
    (void)in_sizes; (void)n_in; (void)out_size; (void)ws_size;
    const float* X        = (const float*)d_in[0];   // (N,4,2)
    const float* enc_Wih  = (const float*)d_in[1];
    const float* enc_Whh  = (const float*)d_in[2];
    const float* enc_bih  = (const float*)d_in[3];
    const float* enc_bhh  = (const float*)d_in[4];
    const float* dec_Wih  = (const float*)d_in[5];
    const float* dec_Whh  = (const float*)d_in[6];
    const float* dec_bih  = (const float*)d_in[7];
    const float* dec_bhh  = (const float*)d_in[8];
    const float* out_W    = (const float*)d_in[9];
    const float* out_b    = (const float*)d_in[10];
    float* out = (float*)d_out;                      // (N,2,2)

    float* hA = (float*)d_ws;
    float* hB = hA + (size_t)GRU_N * GRU_H;          // 2 x 64 MB ping-pong

    dim3 blk(256);
    dim3 gemm_grid(GRU_N / BM, GRU_H / BN);          // 256 x 16

    // encoder t=0 (h==0 shortcut), x = X[:,0,:]
    gru_step0_kernel<<<(GRU_N * GRU_H) / 256, blk, 0, stream>>>(
        hA, X, enc_Wih, enc_bih, enc_bhh);
    // encoder t=1, x = X[:,1,:]
    gru_step_kernel<<<gemm_grid, blk, 0, stream>>>(
        hA, hB, X + 2, 8, enc_Whh, enc_Wih, enc_bih, enc_bhh);
    // decoder t=0, x = X[:,1,:]
    gru_step_kernel<<<gemm_grid, blk, 0, stream>>>(
        hB, hA, X + 2, 8, dec_Whh, dec_Wih, dec_bih, dec_bhh);
    proj_kernel<<<GRU_N / 8, blk, 0, stream>>>(hA, out_W, out_b, out + 0);
    // decoder t=1, x = outs[0] = d_out[:,0,:]
    gru_step_kernel<<<gemm_grid, blk, 0, stream>>>(
        hA, hB, out + 0, 4, dec_Whh, dec_Wih, dec_bih, dec_bhh);
    proj_kernel<<<GRU_N / 8, blk, 0, stream>>>(hB, out_W, out_b, out + 2);
}